// ParticleLossModule_39247411151191
// MI455X (gfx1250) — compile-verified
//
#include <hip/hip_runtime.h>
#include <hip/hip_bf16.h>

typedef __attribute__((ext_vector_type(2))) float v2f;
typedef __attribute__((ext_vector_type(8))) float v8f;

// ---------------- constants (folded in double, cast to f32 like JAX) --------
#define N_GRID   128
#define GCELLS   2097152           // 128^3
constexpr double dDX  = 1.0 / 128.0;
constexpr double dDT  = 5e-05;
constexpr double dPV  = dDX * dDX * dDX;           // P_VOL
constexpr double dPM  = 1.0 * dPV;                 // P_MASS
constexpr double dLA  = 1200.0 * 0.3 / ((1.0 + 0.3) * (1.0 - 2.0 * 0.3));
constexpr double dSK  = -dDT * dPV * 4.0 * 128.0 * 128.0 * dLA; // stress scale (MU==0)
#define DXF      ((float)dDX)
#define PMF      ((float)dPM)
#define SKF      ((float)dSK)
#define EPSF     (1e-14f)
#define LSCALE   (1e11f)

__device__ __forceinline__ void atomAddF(float* p, float v) {
  __hip_atomic_fetch_add(p, v, __ATOMIC_RELAXED, __HIP_MEMORY_SCOPE_AGENT);
}

// ---------------- P2G scatter: one thread per particle ----------------------
// grid layout in ws: m_pre[G], m_tru[G], dvx[G], dvy[G], dvz[G]
__global__ void p2g_kernel(const float* __restrict__ phys,
                           const float* __restrict__ mask,
                           float* __restrict__ m_arr,   // m_pre or m_tru
                           float* __restrict__ dv,      // base of dv[3*G]
                           float sign, int N) {
  int p = blockIdx.x * blockDim.x + threadIdx.x;
  if (p >= N) return;

  // 24 floats per particle; 96B stride -> always 16B aligned
  const float4* q = (const float4*)(phys + (size_t)p * 24);
  float4 q0 = q[0], q1 = q[1], q2 = q[2], q3 = q[3], q4 = q[4], q5 = q[5];
  float x0 = q0.x, x1 = q0.y, x2 = q0.z;
  float v0 = q0.w, v1 = q1.x, v2 = q1.y;
  float C00 = q1.z, C01 = q1.w, C02 = q2.x;
  float C10 = q2.y, C11 = q2.z, C12 = q2.w;
  float C20 = q3.x, C21 = q3.y, C22 = q3.z;
  float F00 = q3.w, F01 = q4.x, F02 = q4.y;
  float F10 = q4.z, F11 = q4.w, F12 = q5.x;
  float F20 = q5.y, F21 = q5.z, F22 = q5.w;

  // clamp position, base cell, fractional offset
  const float lo = 0.5f * DXF, hi = 1.0f - 0.5f * DXF;
  x0 = fminf(fmaxf(x0, lo), hi);
  x1 = fminf(fmaxf(x1, lo), hi);
  x2 = fminf(fmaxf(x2, lo), hi);
  float s0 = x0 * (float)N_GRID, s1 = x1 * (float)N_GRID, s2 = x2 * (float)N_GRID;
  float b0f = floorf(s0 - 0.5f), b1f = floorf(s1 - 0.5f), b2f = floorf(s2 - 0.5f);
  int bx = (int)b0f, by = (int)b1f, bz = (int)b2f;
  float fx0 = s0 - b0f, fx1 = s1 - b1f, fx2 = s2 - b2f;

  // quadratic B-spline weights per dim
  float wx[3], wy[3], wz[3];
  { float a;
    a = 1.5f - fx0; wx[0] = 0.5f * a * a; a = fx0 - 1.0f; wx[1] = 0.75f - a * a; a = fx0 - 0.5f; wx[2] = 0.5f * a * a;
    a = 1.5f - fx1; wy[0] = 0.5f * a * a; a = fx1 - 1.0f; wy[1] = 0.75f - a * a; a = fx1 - 0.5f; wy[2] = 0.5f * a * a;
    a = 1.5f - fx2; wz[0] = 0.5f * a * a; a = fx2 - 1.0f; wz[1] = 0.75f - a * a; a = fx2 - 0.5f; wz[2] = 0.5f * a * a; }

  // J = |det F|; MU==0 -> stress = s*I with s = SKF * J * (J-1)
  float det = F00 * (F11 * F22 - F12 * F21)
            - F01 * (F10 * F22 - F12 * F20)
            + F02 * (F10 * F21 - F11 * F20);
  float J = fabsf(det);
  float sp = SKF * J * (J - 1.0f);

  // affine = s*I + P_MASS * C
  float a00 = PMF * C00 + sp, a01 = PMF * C01,      a02 = PMF * C02;
  float a10 = PMF * C10,      a11 = PMF * C11 + sp, a12 = PMF * C12;
  float a20 = PMF * C20,      a21 = PMF * C21,      a22 = PMF * C22 + sp;

  float mk = mask[p];
  float* dvx = dv;
  float* dvy = dv + (size_t)GCELLS;
  float* dvz = dv + 2 * (size_t)GCELLS;

  #pragma unroll
  for (int i = 0; i < 3; ++i) {
    int gx = bx + i;
    float wi = wx[i];
    float dpx = ((float)i - fx0) * DXF;
    #pragma unroll
    for (int j = 0; j < 3; ++j) {
      int gy = by + j;
      float wij = wi * wy[j];
      float dpy = ((float)j - fx1) * DXF;
      #pragma unroll
      for (int k = 0; k < 3; ++k) {
        int gz = bz + k;
        float wt = wij * wz[k] * mk;
        float dpz = ((float)k - fx2) * DXF;
        // OOB cells can carry nonzero weight; their contributions are dropped
        if ((unsigned)gx < (unsigned)N_GRID &&
            (unsigned)gy < (unsigned)N_GRID &&
            (unsigned)gz < (unsigned)N_GRID) {
          int cell = (gx << 14) + (gy << 7) + gz;
          float m0 = wt * (PMF * v0 + a00 * dpx + a01 * dpy + a02 * dpz);
          float m1 = wt * (PMF * v1 + a10 * dpx + a11 * dpy + a12 * dpz);
          float m2 = wt * (PMF * v2 + a20 * dpx + a21 * dpy + a22 * dpz);
          atomAddF(m_arr + cell, wt * PMF);
          atomAddF(dvx + cell, sign * m0);
          atomAddF(dvy + cell, sign * m1);
          atomAddF(dvz + cell, sign * m2);
        }
      }
    }
  }
}

// ---------------- grid reduction with V_WMMA_F32_16X16X4_F32 ----------------
// 512 blocks x 256 threads x 2 cells/lane x 8 iters == 128^3 exactly: no
// divergence, EXEC all-ones (WMMA requirement).
#define RED_BLOCKS 512
#define RED_THREADS 256
#define RED_LANES (RED_BLOCKS * RED_THREADS)   // 131072
#define RED_ITERS (GCELLS / (2 * RED_LANES))   // 8

__global__ void reduce_kernel(const float* __restrict__ grid,
                              float* __restrict__ acc) {
  const float* m_pre = grid;
  const float* m_tru = grid + (size_t)GCELLS;
  const float* dvx   = grid + 2 * (size_t)GCELLS;
  const float* dvy   = grid + 3 * (size_t)GCELLS;
  const float* dvz   = grid + 4 * (size_t)GCELLS;

  unsigned g = blockIdx.x * RED_THREADS + threadIdx.x;
  v8f c_lm = {}; v8f c_lv = {}; v8f c_cn = {};
  v2f ones; ones[0] = 1.0f; ones[1] = 1.0f;

  for (int it = 0; it < RED_ITERS; ++it) {
    unsigned c0 = (unsigned)it * (2 * RED_LANES) + g;
    unsigned c1 = c0 + RED_LANES;

    float mp0 = m_pre[c0], mt0 = m_tru[c0];
    float mp1 = m_pre[c1], mt1 = m_tru[c1];
    float x0 = dvx[c0], y0 = dvy[c0], z0 = dvz[c0];
    float x1 = dvx[c1], y1 = dvy[c1], z1 = dvz[c1];

    bool act0 = (mp0 > EPSF) || (mt0 > EPSF);
    bool act1 = (mp1 > EPSF) || (mt1 > EPSF);
    float dm0 = mp0 - mt0, dm1 = mp1 - mt1;

    v2f a_lm, a_lv, a_cn;
    a_lm[0] = act0 ? dm0 * dm0 : 0.0f;
    a_lm[1] = act1 ? dm1 * dm1 : 0.0f;
    a_lv[0] = act0 ? (x0 * x0 + y0 * y0 + z0 * z0) : 0.0f;
    a_lv[1] = act1 ? (x1 * x1 + y1 * y1 + z1 * z1) : 0.0f;
    a_cn[0] = act0 ? 1.0f : 0.0f;
    a_cn[1] = act1 ? 1.0f : 0.0f;

    // D = A * ones(4x16) + C : accumulates row sums; one WMMA retires 64 adds
    c_lm = __builtin_amdgcn_wmma_f32_16x16x4_f32(false, a_lm, false, ones,
                                                 (short)0, c_lm, false, false);
    c_lv = __builtin_amdgcn_wmma_f32_16x16x4_f32(false, a_lv, false, ones,
                                                 (short)0, c_lv, false, false);
    c_cn = __builtin_amdgcn_wmma_f32_16x16x4_f32(false, a_cn, false, ones,
                                                 (short)0, c_cn, false, false);
  }

  // Each lane's column holds rowsums for its half (lanes 0-15: M=0..7,
  // lanes 16-31: M=8..15). Sum own 8, combine halves via xor-16 shuffle.
  float s_lm = 0.f, s_lv = 0.f, s_cn = 0.f;
  #pragma unroll
  for (int r = 0; r < 8; ++r) { s_lm += c_lm[r]; s_lv += c_lv[r]; s_cn += c_cn[r]; }
  s_lm += __shfl_xor(s_lm, 16, 32);
  s_lv += __shfl_xor(s_lv, 16, 32);
  s_cn += __shfl_xor(s_cn, 16, 32);

  if ((threadIdx.x & 31) == 0) {
    atomAddF(acc + 0, s_lm);
    atomAddF(acc + 1, s_lv);
    atomAddF(acc + 2, s_cn);
  }
}

// ---------------- finalize: per-batch scalar loss ---------------------------
__global__ void finalize_kernel(const float* __restrict__ acc,
                                float* __restrict__ out) {
  int b = threadIdx.x;
  if (b < 2) {
    float lm = acc[b * 4 + 0];
    float lv = acc[b * 4 + 1];
    float cn = acc[b * 4 + 2];
    out[b] = LSCALE * lm / fmaxf(cn, 1.0f) + LSCALE * lv / fmaxf(cn, EPSF);
  }
}

extern "C" void kernel_launch(void* const* d_in, const int* in_sizes, int n_in,
                              void* d_out, int out_size, void* d_ws, size_t ws_size,
                              hipStream_t stream) {
  const float* x_pred = (const float*)d_in[0];
  const float* x_true = (const float*)d_in[1];
  const float* masks  = (const float*)d_in[2];
  // d_in[3]/d_in[4] (R_pred/R_true) are dead: MU == 0 kills the elastic term.

  const int B = 2;
  const int N = in_sizes[2] / B;          // masks is (B, N)

  float* grid = (float*)d_ws;             // 5*G floats, reused per batch
  float* acc  = grid + 5 * (size_t)GCELLS; // 8 floats: [b*4 + {lm,lv,cells}]

  hipMemsetAsync(acc, 0, 8 * sizeof(float), stream);

  dim3 blk(256), grd((N + 255) / 256);
  for (int b = 0; b < B; ++b) {
    hipMemsetAsync(grid, 0, 5 * (size_t)GCELLS * sizeof(float), stream);
    p2g_kernel<<<grd, blk, 0, stream>>>(x_pred + (size_t)b * N * 24,
                                        masks + (size_t)b * N,
                                        grid,                       // m_pre
                                        grid + 2 * (size_t)GCELLS,  // dv base
                                        +1.0f, N);
    p2g_kernel<<<grd, blk, 0, stream>>>(x_true + (size_t)b * N * 24,
                                        masks + (size_t)b * N,
                                        grid + (size_t)GCELLS,      // m_tru
                                        grid + 2 * (size_t)GCELLS,
                                        -1.0f, N);
    reduce_kernel<<<RED_BLOCKS, RED_THREADS, 0, stream>>>(grid, acc + b * 4);
  }
  finalize_kernel<<<1, 32, 0, stream>>>(acc, (float*)d_out);
}